// RPGVoxelGridSurface_46626164966021
// MI455X (gfx1250) — compile-verified
//
#include <hip/hip_runtime.h>

// True clang ext_vector so __builtin_nontemporal_load accepts it
typedef float __attribute__((ext_vector_type(4))) f32x4;

#define BATCH    8
#define NEV      1000000
#define HH       480
#define WW       640
#define NBINS    5
#define PLANE    (WW * HH)          // 307200 elements per time bin
#define GRID_PB  (NBINS * PLANE)    // 1,536,000 elements per batch

// ---------------------------------------------------------------------------
// Hot kernel FIRST in the module so the disasm snippet shows it.
// Bilinear-in-time scatter of events into the voxel grid.
// One float4 (t,x,y,p) per event -> up to two f32 atomic adds.
// Output grid (49 MB total) stays resident in the 192 MB L2 -> atomics
// resolve at L2 atomic units, not HBM.
// ---------------------------------------------------------------------------
__global__ __launch_bounds__(256) void voxel(const float* __restrict__ ev,
                                             float* __restrict__ out) {
    const int b = blockIdx.y;
    const float* __restrict__ evb  = ev  + (size_t)b * NEV * 4;
    float*       __restrict__ outb = out + (size_t)b * GRID_PB;

    // Per-batch uniform scale: (BINS-1) / (t_last - t_first). Two cached loads.
    float t0 = evb[0];
    float tN = evb[(size_t)(NEV - 1) * 4];
    float dT = tN - t0;
    dT = (dT == 0.0f) ? 1.0f : dT;
    const float scale = (float)(NBINS - 1) / dT;

    const f32x4* __restrict__ ev4 = (const f32x4*)evb;
    const int stride = blockDim.x * gridDim.x;

    for (int i = blockIdx.x * blockDim.x + threadIdx.x; i < NEV; i += stride) {
        // Streaming read: each event used exactly once -> NT 128-bit load.
        f32x4 e = __builtin_nontemporal_load(&ev4[i]);

        // Prefetch next grid-stride iteration's event (global_prefetch_b8).
        if (i + stride < NEV)
            __builtin_prefetch((const void*)&ev4[i + stride], 0, 0);

        float ts  = (e.x - t0) * scale;        // in [0, BINS-1], t sorted
        float tis = floorf(ts);
        float dts = ts - tis;
        int   til = (int)tis;

        float pol = (e.w == 0.0f) ? -1.0f : e.w;   // p in {0,1} -> {-1,+1}
        int   xi  = (int)e.y;
        int   yi  = (int)e.z;
        int   base = xi + yi * WW + til * PLANE;

        // Left bin: valid when 0 <= til < BINS
        if (til >= 0 && til < NBINS) {
            (void)__hip_atomic_fetch_add(&outb[base], pol * (1.0f - dts),
                                         __ATOMIC_RELAXED, __HIP_MEMORY_SCOPE_AGENT);
        }
        // Right bin: valid when 0 <= til and til+1 < BINS
        if (til >= 0 && til + 1 < NBINS) {
            (void)__hip_atomic_fetch_add(&outb[base + PLANE], pol * dts,
                                         __ATOMIC_RELAXED, __HIP_MEMORY_SCOPE_AGENT);
        }
    }
}

// ---------------------------------------------------------------------------
// Zero the (poisoned) output grid with wide 128-bit stores.
// ---------------------------------------------------------------------------
__global__ __launch_bounds__(256) void zero_out(f32x4* __restrict__ out, int n4) {
    int i = blockIdx.x * blockDim.x + threadIdx.x;
    if (i < n4) {
        f32x4 z = (f32x4)(0.0f);
        out[i] = z;  // global_store_b128
    }
}

// ---------------------------------------------------------------------------
extern "C" void kernel_launch(void* const* d_in, const int* in_sizes, int n_in,
                              void* d_out, int out_size, void* d_ws, size_t ws_size,
                              hipStream_t stream) {
    const float* ev  = (const float*)d_in[0];   // (B, N, 4) float32
    float*       out = (float*)d_out;           // (B, BINS, H, W) float32

    // 1) zero the output (harness poisons it with 0xAA)
    int n4 = out_size / 4;                      // 3,072,000 float4s
    zero_out<<<(n4 + 255) / 256, 256, 0, stream>>>((f32x4*)out, n4);

    // 2) scatter events
    dim3 grid(512, BATCH);                      // 4096 blocks, 32768 wave32s
    voxel<<<grid, 256, 0, stream>>>(ev, out);
}